// EpisodicMemorySystem_70068096467321
// MI455X (gfx1250) — compile-verified
//
#include <hip/hip_runtime.h>
#include <math.h>

#define C_DIM 8
#define M_DIM 8192
#define D_DIM 128
#define B_DIM 32
#define K_TOP 16
#define EPSF  1e-8f

typedef __attribute__((ext_vector_type(2))) float v2f;
typedef __attribute__((ext_vector_type(8))) float v8f;
typedef __attribute__((ext_vector_type(4))) unsigned int u32x4;
typedef __attribute__((ext_vector_type(8))) unsigned int u32x8;

// ---------------------------------------------------------------------------
// Tensor Data Mover: DMA a 2D fp32 tile (tile_d1 rows x tile_d0 elems, row
// stride `stride` elems) from global to LDS.  Optional LDS row padding via the
// TDM pad feature: pad_iv = pad_interval code (6 -> every 128 DWORDs),
// pad_dw = pad amount in DWORDs (0 = no padding).
// Issue from ONE wave only (EXEC is ignored; each executing wave issues a DMA).
// ---------------------------------------------------------------------------
__device__ __forceinline__ void tdm_tile_load(const float* gsrc, void* lds_dst,
                                              unsigned tile_d0, unsigned tile_d1,
                                              unsigned stride, unsigned pad_iv,
                                              unsigned pad_dw) {
#if defined(__HIP_DEVICE_COMPILE__)
  unsigned long long ga = (unsigned long long)(uintptr_t)gsrc;
  unsigned lds = (unsigned)(uintptr_t)lds_dst;   // low 32 bits = LDS byte offset
  u32x4 g0;
  g0[0] = 1u;                                    // count=1, user descriptor
  g0[1] = lds;                                   // lds_addr
  g0[2] = (unsigned)(ga & 0xFFFFFFFFu);          // global_addr[31:0]
  g0[3] = (unsigned)((ga >> 32) & 0x1FFFFFFu) | (2u << 30);  // addr[56:32]|type=2
  unsigned w0 = (2u << 16);                      // data_size = 4 bytes
  if (pad_dw)
    w0 |= (1u << 20) | (pad_iv << 22) | ((pad_dw - 1u) << 25);
  u32x8 g1;
  g1[0] = w0;
  g1[1] = (tile_d0 & 0xFFFFu) << 16;                               // tensor_dim0 lo
  g1[2] = ((tile_d0 >> 16) & 0xFFFFu) | ((tile_d1 & 0xFFFFu) << 16); // d0 hi | tensor_dim1 lo
  g1[3] = ((tile_d1 >> 16) & 0xFFFFu) | ((tile_d0 & 0xFFFFu) << 16); // d1 hi | tile_dim0
  g1[4] = (tile_d1 & 0xFFFFu);                                     // tile_dim1 (tile_dim2=0)
  g1[5] = stride;                                                  // tensor_dim0_stride lo32
  g1[6] = 0u;
  g1[7] = 0u;
  asm volatile("tensor_load_to_lds %0, %1" :: "s"(g0), "s"(g1) : "memory");
#else
  (void)gsrc; (void)lds_dst; (void)tile_d0; (void)tile_d1;
  (void)stride; (void)pad_iv; (void)pad_dw;
#endif
}

__device__ __forceinline__ void wait_tensorcnt0() {
#if defined(__HIP_DEVICE_COMPILE__)
  __builtin_amdgcn_s_wait_tensorcnt(0);
#endif
}

// ---------------------------------------------------------------------------
// K1: hqT[c][e][b] = sum_d q[b][d] * W1[c][e][d]  + b1[c][e]
// ---------------------------------------------------------------------------
__global__ __launch_bounds__(128)
void hq_kernel(const float* __restrict__ q, const float* __restrict__ W1,
               const float* __restrict__ b1, float* __restrict__ hqT) {
  __shared__ float s_q[D_DIM];
  int bid = blockIdx.x;                // c*B + b
  int c = bid / B_DIM, b = bid % B_DIM;
  int e = threadIdx.x;                 // 0..127
  s_q[e] = q[b * D_DIM + e];
  __syncthreads();
  const float* wrow = W1 + (size_t)(c * D_DIM + e) * (2 * D_DIM); // Wq row
  float acc = 0.f;
#pragma unroll 8
  for (int d = 0; d < D_DIM; ++d) acc += s_q[d] * wrow[d];
  hqT[(c * D_DIM + e) * B_DIM + b] = acc + b1[c * D_DIM + e];
}

// ---------------------------------------------------------------------------
// K2: qn[b][d] = qc[b][d] / max(||qc[b]||, eps)   (wave per row, wave32)
// ---------------------------------------------------------------------------
__global__ __launch_bounds__(1024)
void qn_kernel(const float* __restrict__ qc, float* __restrict__ qn) {
  int t = threadIdx.x;
  int b = t >> 5, lane = t & 31;
  float v[4]; float ss = 0.f;
#pragma unroll
  for (int j = 0; j < 4; ++j) { v[j] = qc[b * D_DIM + lane + 32 * j]; ss += v[j] * v[j]; }
#pragma unroll
  for (int m = 16; m >= 1; m >>= 1) ss += __shfl_xor(ss, m, 32);
  float inv = 1.f / fmaxf(sqrtf(ss), EPSF);
#pragma unroll
  for (int j = 0; j < 4; ++j) qn[b * D_DIM + lane + 32 * j] = v[j] * inv;
}

// ---------------------------------------------------------------------------
// K3: hm[c][m][e] = sum_d mem[c][m][d] * W1[c][e][D+d]   (WMMA f32 16x16x4)
// Content tile arrives via TDM (padded rows); WmT stored pair-interleaved so
// each B fragment is a single aligned ds_load_b64.
// ---------------------------------------------------------------------------
__global__ __launch_bounds__(256)
void hm_wmma_kernel(const float* __restrict__ mem, const float* __restrict__ W1,
                    float* __restrict__ hm) {
  __shared__ float s_wm[D_DIM * D_DIM];  // WmT pair-interleaved: [(d>>1)][e*2+(d&1)]
  __shared__ float s_a[128 * 132];       // content tile [m][d], stride 132 (TDM pad)
  int c = blockIdx.y;
  int mbase = blockIdx.x * 128;
  int tid = threadIdx.x;

  if (tid < 32)   // async DMA of the 128x128 content tile, +16B pad per row
    tdm_tile_load(mem + ((size_t)c * M_DIM + mbase) * D_DIM, s_a,
                  128u, 128u, 128u, 6u, 4u);

  for (int i = tid; i < D_DIM * D_DIM; i += 256) {   // overlap: WmT transpose
    int e = i >> 7, d = i & 127;
    s_wm[(d >> 1) * 256 + e * 2 + (d & 1)] =
        W1[(size_t)(c * D_DIM + e) * 256 + D_DIM + d];
  }
  if (tid < 32) wait_tensorcnt0();
  __syncthreads();

  int wave = tid >> 5, lane = tid & 31;
  int half = lane >> 4, l16 = lane & 15;
  int mt = wave;
  v8f acc[8] = {};
  int arow = (mt * 16 + l16) * 132 + half * 2;
  for (int k0 = 0; k0 < D_DIM; k0 += 4) {
    v2f a = *(const v2f*)&s_a[arow + k0];
    int bbase = ((k0 + half * 2) >> 1) * 256 + l16 * 2;
#pragma unroll
    for (int et = 0; et < 8; ++et) {
      v2f bf = *(const v2f*)&s_wm[bbase + et * 32];
      acc[et] = __builtin_amdgcn_wmma_f32_16x16x4_f32(false, a, false, bf,
                                                      (short)0, acc[et], false, false);
    }
  }
  float* out = hm + ((size_t)c * M_DIM + mbase) * D_DIM;
#pragma unroll
  for (int et = 0; et < 8; ++et)
#pragma unroll
    for (int r = 0; r < 8; ++r) {
      int m = mt * 16 + half * 8 + r;
      int e = et * 16 + l16;
      out[m * D_DIM + e] = acc[et][r];
    }
}

// ---------------------------------------------------------------------------
// K4: ctxws[c][m][b] = (qn[b] . mctx[c][m]) / max(||mctx[c][m]||, eps)
// ---------------------------------------------------------------------------
__global__ __launch_bounds__(256)
void ctx_wmma_kernel(const float* __restrict__ mctx, const float* __restrict__ qn,
                     float* __restrict__ ctxws) {
  __shared__ float s_t[128 * 132];       // contexts tile [m][d] (TDM pad)
  __shared__ float s_qn[B_DIM * 132];    // qn [b][d]        (TDM pad)
  __shared__ float s_minv[128];
  int c = blockIdx.y;
  int mbase = blockIdx.x * 128;
  int tid = threadIdx.x;

  if (tid < 32) {
    tdm_tile_load(mctx + ((size_t)c * M_DIM + mbase) * D_DIM, s_t,
                  128u, 128u, 128u, 6u, 4u);
    tdm_tile_load(qn, s_qn, 128u, 32u, 128u, 6u, 4u);
    wait_tensorcnt0();
  }
  __syncthreads();

  int wave = tid >> 5, lane = tid & 31;
  for (int r = 0; r < 16; ++r) {          // wave w -> rows [w*16, w*16+16)
    int m = wave * 16 + r;
    float ss = 0.f;
#pragma unroll
    for (int j = 0; j < 4; ++j) { float v = s_t[m * 132 + lane + 32 * j]; ss += v * v; }
#pragma unroll
    for (int sh = 16; sh >= 1; sh >>= 1) ss += __shfl_xor(ss, sh, 32);
    if (lane == 0) s_minv[m] = 1.f / fmaxf(sqrtf(ss), EPSF);
  }
  __syncthreads();

  int half = lane >> 4, l16 = lane & 15;
  int mt = wave;                          // N tile (m)
  v8f acc[2] = {};
  int mrow = (mt * 16 + l16) * 132 + half * 2;
  for (int k0 = 0; k0 < D_DIM; k0 += 4) {
    v2f bf = *(const v2f*)&s_t[mrow + k0];     // B element [k][m] = s_t[m][k]
#pragma unroll
    for (int bt = 0; bt < 2; ++bt) {
      v2f a = *(const v2f*)&s_qn[(bt * 16 + l16) * 132 + half * 2 + k0];
      acc[bt] = __builtin_amdgcn_wmma_f32_16x16x4_f32(false, a, false, bf,
                                                      (short)0, acc[bt], false, false);
    }
  }
  float minv = s_minv[mt * 16 + l16];
  float* out = ctxws + ((size_t)c * M_DIM + mbase) * B_DIM;
#pragma unroll
  for (int bt = 0; bt < 2; ++bt)
#pragma unroll
    for (int r = 0; r < 8; ++r) {
      int b = bt * 16 + half * 8 + r;
      int m = mt * 16 + l16;
      out[m * B_DIM + b] = acc[bt][r] * minv;
    }
}

// ---------------------------------------------------------------------------
// K5: final_sim[c][b][m] = 0.5*sigmoid(sum_d relu(hqT[e][b]+hm[m][e])*W2 + b2)
//                          + 0.3*ctx + 0.2*fresh[m]
// ---------------------------------------------------------------------------
__global__ __launch_bounds__(256)
void score_kernel(const float* __restrict__ hm, const float* __restrict__ hqT,
                  const float* __restrict__ ctxws, const float* __restrict__ W2,
                  const float* __restrict__ b2, const float* __restrict__ fresh,
                  float* __restrict__ fs) {
  __shared__ float s_hm[128 * D_DIM];
  __shared__ float s_hq[D_DIM * B_DIM];
  __shared__ float s_w2[D_DIM];
  __shared__ float s_fr[128];
  __shared__ float s_f[128 * 33];
  int c = blockIdx.y;
  int mbase = blockIdx.x * 128;
  int tid = threadIdx.x;

  if (tid < 32) {   // 1-D TDM DMAs: hm tile (64KB) and hqT slice (16KB)
    tdm_tile_load(hm + ((size_t)c * M_DIM + mbase) * D_DIM, s_hm,
                  16384u, 1u, 16384u, 0u, 0u);
    tdm_tile_load(hqT + (size_t)c * D_DIM * B_DIM, s_hq,
                  4096u, 1u, 4096u, 0u, 0u);
  }
  if (tid < D_DIM) s_w2[tid] = W2[c * D_DIM + tid];
  if (tid < 128)   s_fr[tid] = fresh[c * M_DIM + mbase + tid];
  if (tid < 32) wait_tensorcnt0();
  __syncthreads();

  int wave = tid >> 5, b = tid & 31;
  float bias2 = b2[c];
  for (int m = wave; m < 128; m += 8) {
    float ctx = ctxws[((size_t)c * M_DIM + mbase + m) * B_DIM + b];
    const float* hmr = &s_hm[m * D_DIM];
    float acc = bias2;
#pragma unroll 4
    for (int d = 0; d < D_DIM; ++d) {
      float x = hmr[d] + s_hq[d * B_DIM + b];
      acc += fmaxf(x, 0.f) * s_w2[d];
    }
    float cs = 1.f / (1.f + __expf(-acc));
    s_f[m * 33 + b] = 0.5f * cs + 0.3f * ctx + 0.2f * s_fr[m];
  }
  __syncthreads();
  for (int i = tid; i < 128 * B_DIM; i += 256) {     // coalesced transpose-out
    int bb = i >> 7, mm = i & 127;
    fs[((size_t)c * B_DIM + bb) * M_DIM + mbase + mm] = s_f[mm * 33 + bb];
  }
}

// ---------------------------------------------------------------------------
// K6: per (c,b) row: top-16 (iterative argmax, lowest-index ties) + gathers
// ---------------------------------------------------------------------------
__global__ __launch_bounds__(256)
void topk_kernel(const float* __restrict__ fs, const float* __restrict__ mem,
                 const float* __restrict__ fresh, float* __restrict__ out) {
  __shared__ float s_v[M_DIM];
  __shared__ float s_rv[256];
  __shared__ int   s_ri[256];
  __shared__ int   s_idx[K_TOP];
  __shared__ float s_val[K_TOP];
  int row = blockIdx.x;                 // c*B + b
  int c = row >> 5;
  int tid = threadIdx.x;
  const float* src = fs + (size_t)row * M_DIM;
  for (int i = tid; i < M_DIM; i += 256) s_v[i] = src[i];
  __syncthreads();
  for (int k = 0; k < K_TOP; ++k) {
    float bv = -INFINITY; int bi = 0;
    for (int i = tid; i < M_DIM; i += 256) {
      float v = s_v[i];
      if (v > bv) { bv = v; bi = i; }
    }
    s_rv[tid] = bv; s_ri[tid] = bi;
    __syncthreads();
    for (int s = 128; s > 0; s >>= 1) {
      if (tid < s) {
        float ov = s_rv[tid + s]; int oi = s_ri[tid + s];
        if (ov > s_rv[tid] || (ov == s_rv[tid] && oi < s_ri[tid])) {
          s_rv[tid] = ov; s_ri[tid] = oi;
        }
      }
      __syncthreads();
    }
    if (tid == 0) { s_idx[k] = s_ri[0]; s_val[k] = s_rv[0]; s_v[s_ri[0]] = -INFINITY; }
    __syncthreads();
  }
  const size_t RC = (size_t)C_DIM * B_DIM * K_TOP * D_DIM;
  float* rc = out;
  float* ts = out + RC;
  float* tw = out + RC + (size_t)C_DIM * B_DIM * K_TOP;
  if (tid < K_TOP) {
    ts[row * K_TOP + tid] = s_val[tid];
    tw[row * K_TOP + tid] = fresh[c * M_DIM + s_idx[tid]];
  }
  for (int i = tid; i < K_TOP * D_DIM; i += 256) {
    int k = i >> 7, d = i & 127;
    rc[((size_t)row * K_TOP + k) * D_DIM + d] = mem[((size_t)c * M_DIM + s_idx[k]) * D_DIM + d];
  }
}

// ---------------------------------------------------------------------------
extern "C" void kernel_launch(void* const* d_in, const int* in_sizes, int n_in,
                              void* d_out, int out_size, void* d_ws, size_t ws_size,
                              hipStream_t stream) {
  const float* q     = (const float*)d_in[0];
  const float* qc    = (const float*)d_in[1];
  const float* mem   = (const float*)d_in[2];
  const float* mctx  = (const float*)d_in[3];
  const float* fresh = (const float*)d_in[4];
  const float* W1    = (const float*)d_in[5];
  const float* b1    = (const float*)d_in[6];
  const float* W2    = (const float*)d_in[7];
  const float* b2    = (const float*)d_in[8];
  (void)in_sizes; (void)n_in; (void)out_size; (void)ws_size;

  float* ws   = (float*)d_ws;
  float* hqT  = ws;                                   // C*D*B      = 32768
  float* qn   = hqT + (size_t)C_DIM * D_DIM * B_DIM;  // B*D        = 4096
  float* hm   = qn + (size_t)B_DIM * D_DIM;           // C*M*D      = 8388608
  float* ctxw = hm + (size_t)C_DIM * M_DIM * D_DIM;   // C*M*B      = 2097152
  float* fs   = ctxw + (size_t)C_DIM * M_DIM * B_DIM; // C*B*M      = 2097152
  float* outp = (float*)d_out;

  hipLaunchKernelGGL(hq_kernel, dim3(C_DIM * B_DIM), dim3(128), 0, stream, q, W1, b1, hqT);
  hipLaunchKernelGGL(qn_kernel, dim3(1), dim3(1024), 0, stream, qc, qn);
  hipLaunchKernelGGL(hm_wmma_kernel, dim3(M_DIM / 128, C_DIM), dim3(256), 0, stream, mem, W1, hm);
  hipLaunchKernelGGL(ctx_wmma_kernel, dim3(M_DIM / 128, C_DIM), dim3(256), 0, stream, mctx, qn, ctxw);
  hipLaunchKernelGGL(score_kernel, dim3(M_DIM / 128, C_DIM), dim3(256), 0, stream,
                     hm, hqT, ctxw, W2, b2, fresh, fs);
  hipLaunchKernelGGL(topk_kernel, dim3(C_DIM * B_DIM), dim3(256), 0, stream, fs, mem, fresh, outp);
}